// Pitch_5179730559556
// MI455X (gfx1250) — compile-verified
//
#include <hip/hip_runtime.h>
#include <stdint.h>

typedef float v2f __attribute__((ext_vector_type(2)));
typedef float v8f __attribute__((ext_vector_type(8)));

#define NSAMP    80000
#define HOP      80
#define TAU_MIN  20
#define TAU_MAX  133
#define WLEN     133      // correlation window length
#define NOUT     996      // output frames per batch row
#define FPB      4        // frames per workgroup (one per wave)
#define SIG_N    516      // samples staged per block: covers max index 80*3+275=515

__device__ __forceinline__ v8f wmma4(v2f a, v2f b, v8f c) {
  // D = A(16x4, f32) * B(4x16, f32) + C(16x16, f32)
  return __builtin_amdgcn_wmma_f32_16x16x4_f32(false, a, false, b, (short)0, c,
                                               false, false);
}

__global__ __launch_bounds__(128) void yin_wmma_kernel(const float* __restrict__ x,
                                                       float* __restrict__ out) {
  __shared__ float sig[SIG_N + 8];       // staged signal window (shared by 4 frames)
  __shared__ float cbuf[FPB][144];       // c(tau) = G[0,tau]
  __shared__ float ebuf[FPB][144];       // E(tau) = G[tau,tau]

  const int tid  = threadIdx.x;
  const int w    = tid >> 5;             // wave id == frame within block
  const int lane = tid & 31;
  const int b    = blockIdx.y;
  const int blk  = blockIdx.x;
  const int frame = blk * FPB + w;

  // ---- stage signal window into LDS via CDNA5 async global->LDS copies ----
  const float* src = x + (size_t)b * NSAMP + (size_t)blk * (HOP * FPB);
  const unsigned ldsBase = (unsigned)(uintptr_t)(&sig[0]); // low 32 bits of generic ptr = LDS offset
  for (int i = tid; i < SIG_N; i += 128) {
    unsigned loff = ldsBase + 4u * (unsigned)i;
    uint64_t ga   = (uint64_t)(uintptr_t)(src + i);
    asm volatile("global_load_async_to_lds_b32 %0, %1, off"
                 :: "v"(loff), "v"(ga) : "memory");
  }
#if __has_builtin(__builtin_amdgcn_s_wait_asynccnt)
  __builtin_amdgcn_s_wait_asynccnt(0);
#else
  asm volatile("s_wait_asynccnt 0" ::: "memory");
#endif
  __syncthreads();

  // ---- per-lane WMMA operand addressing (A and B share the same pattern) ----
  const int half  = lane >> 4;           // 0: K=0,1  1: K=2,3
  const int m     = lane & 15;           // M (for A) == N (for B)
  const int fb    = HOP * w;             // frame base within sig
  const int lbase = fb + m + 2 * half;

  for (int t = 0; t < 9; ++t) {
    const int S = 16 * t;
    v8f accD = {0.f, 0.f, 0.f, 0.f, 0.f, 0.f, 0.f, 0.f};  // diag tile (S,S)
    v8f accR = {0.f, 0.f, 0.f, 0.f, 0.f, 0.f, 0.f, 0.f};  // row  tile (0,S)
    const float* pS = &sig[lbase + S];
    const float* p0 = &sig[lbase];

    // 33 full K-chunks (k = 0..131, all in-range)
    for (int c = 0; c < 132; c += 4) {
      v2f vs; vs.x = pS[c]; vs.y = pS[c + 1];
      accD = wmma4(vs, vs, accD);
      if (t > 0) {
        v2f v0; v0.x = p0[c]; v0.y = p0[c + 1];
        accR = wmma4(v0, vs, accR);
      }
    }
    // tail chunk c=132: mask k >= 133
    {
      const int c  = 132;
      const int kk = c + 2 * half;
      v2f vs;
      vs.x = (kk     < WLEN) ? pS[c]     : 0.0f;
      vs.y = (kk + 1 < WLEN) ? pS[c + 1] : 0.0f;
      accD = wmma4(vs, vs, accD);
      if (t > 0) {
        v2f v0;
        v0.x = (kk     < WLEN) ? p0[c]     : 0.0f;
        v0.y = (kk + 1 < WLEN) ? p0[c + 1] : 0.0f;
        accR = wmma4(v0, vs, accR);
      }
    }

    // extract diagonal D[i,i] -> E(S+i)
    // i<8:  VGPR i,   lane i       ; i>=8: VGPR i-8, lane i+16
    if (lane < 8)        ebuf[w][S + lane]      = accD[lane];
    else if (lane >= 24) ebuf[w][S + lane - 16] = accD[lane - 24];
    // extract row 0: D[0,n] -> c(S+n)   (VGPR 0, lanes 0..15)
    if (t == 0) {
      if (lane < 16) cbuf[w][lane] = accD[0];
    } else {
      if (lane < 16) cbuf[w][S + lane] = accR[0];
    }
  }

  __syncthreads();

  // ---- CMNDF + threshold pick (serial per frame, 133 steps: negligible) ----
  if (lane == 0) {
    const float E0 = ebuf[w][0];
    float cum = 0.0f;
    int tauHit = 0;
    for (int tau = 1; tau <= TAU_MAX; ++tau) {
      float d = E0 + ebuf[w][tau] - 2.0f * cbuf[w][tau];
      cum += d;
      float cm = d * (float)tau / fmaxf(cum, 1e-8f);
      if (tau >= TAU_MIN && cm < 0.2f) { tauHit = tau; break; }
    }
    out[(size_t)b * NOUT + frame] = tauHit ? (8000.0f / (float)tauHit) : 0.0f;
  }
}

extern "C" void kernel_launch(void* const* d_in, const int* in_sizes, int n_in,
                              void* d_out, int out_size, void* d_ws, size_t ws_size,
                              hipStream_t stream) {
  const float* x = (const float*)d_in[0];
  float* out = (float*)d_out;
  const int B = in_sizes[0] / NSAMP;     // 8
  dim3 grid(NOUT / FPB, B);              // 249 x 8 workgroups
  dim3 block(128);                       // 4 waves -> 4 frames per workgroup
  yin_wmma_kernel<<<grid, block, 0, stream>>>(x, out);
}